// GAT_87780541595669
// MI455X (gfx1250) — compile-verified
//
#include <hip/hip_runtime.h>
#include <hip/hip_bf16.h>

typedef __attribute__((ext_vector_type(2))) float v2f;
typedef __attribute__((ext_vector_type(8))) float v8f;

#ifndef NEG_SLOPE
#define NEG_SLOPE 0.2f
#endif

// ---------- helpers ----------
__device__ __forceinline__ unsigned fenc(float f) {
    unsigned u = __float_as_uint(f);
    return (u & 0x80000000u) ? ~u : (u | 0x80000000u);
}
__device__ __forceinline__ float fdec(unsigned u) {
    u = (u & 0x80000000u) ? (u & 0x7FFFFFFFu) : ~u;
    return __uint_as_float(u);
}

// ---------- init: zero accumulators ----------
__global__ void k_init(float* out, float* agg, unsigned* mu, float* ssum,
                       float* scal, long long n64, int n) {
    long long i = (long long)blockIdx.x * blockDim.x + threadIdx.x;
    if (i < n64) { out[i] = 0.0f; agg[i] = 0.0f; }
    if (i < n)   { mu[i] = 0u; ssum[i] = 0.0f; }
    if (i < 8)   { scal[i] = 0.0f; }
}

// ---------- small reductions: mean(edge_attr), ce1=We1.att_e1, ce2=We2.att_e2 ----------
__global__ void k_reduce(const float* __restrict__ ea, int E,
                         const float* __restrict__ We1, const float* __restrict__ ae1,
                         const float* __restrict__ We2, const float* __restrict__ ae2,
                         float* scal) {
    __shared__ float sh[256];
    float part = 0.0f;
    for (long long i = (long long)blockIdx.x * blockDim.x + threadIdx.x; i < E;
         i += (long long)gridDim.x * blockDim.x)
        part += ea[i];
    sh[threadIdx.x] = part;
    __syncthreads();
    for (int s = 128; s > 0; s >>= 1) {
        if (threadIdx.x < s) sh[threadIdx.x] += sh[threadIdx.x + s];
        __syncthreads();
    }
    if (threadIdx.x == 0) atomicAdd(&scal[0], sh[0]);
    if (blockIdx.x == 0 && threadIdx.x < 64) {
        atomicAdd(&scal[1], We1[threadIdx.x] * ae1[threadIdx.x]);
        atomicAdd(&scal[2], We2[threadIdx.x] * ae2[threadIdx.x]);
    }
}

// ---------- WMMA GEMM: C[nrows,64] = A[nrows,64] @ W[64,64] (+bias) ----------
// One 16x16 output tile per wave; K=64 via 16 chained V_WMMA_F32_16X16X4_F32.
template <bool ADD_BIAS>
__global__ __launch_bounds__(256) void k_gemm64(const float* __restrict__ A,
                                                const float* __restrict__ W,
                                                const float* __restrict__ bias,
                                                float* __restrict__ C, int nrows) {
    const int lane = threadIdx.x & 31;
    const int wave = threadIdx.x >> 5;
    const int row_tile = blockIdx.x * 2 + (wave >> 2);
    const int col_tile = wave & 3;
    if (row_tile * 16 >= nrows) return;          // uniform per-wave exit, EXEC stays full

    const int half = lane >> 4;                  // 0: K pair {0,1}, 1: K pair {2,3}
    const int l15  = lane & 15;
    int row = row_tile * 16 + l15;
    if (row >= nrows) row = nrows - 1;           // clamp reads; stores guarded below
    const int col = col_tile * 16 + l15;

    const float* Arow = A + (long long)row * 64 + half * 2;
    const float* Wc   = W + col + half * 2 * 64;

    v8f acc = {};
#pragma unroll
    for (int kb = 0; kb < 16; ++kb) {
        v2f a = *(const v2f*)(Arow + kb * 4);
        v2f b;
        b.x = Wc[kb * 256];
        b.y = Wc[kb * 256 + 64];
        acc = __builtin_amdgcn_wmma_f32_16x16x4_f32(
            /*neg_a=*/false, a, /*neg_b=*/false, b,
            /*c_mod=*/(short)0, acc, /*reuse_a=*/false, /*reuse_b=*/false);
    }
#pragma unroll
    for (int r = 0; r < 8; ++r) {
        int orow = row_tile * 16 + r + half * 8;
        if (orow < nrows) {
            float v = acc[r];
            if (ADD_BIAS) v += bias[col];
            C[(long long)orow * 64 + col] = v;
        }
    }
}

// ---------- per-node attention coefficients: a_src[i]=xl[i].att_src, a_dst[i]=xl[i].att_dst ----------
__global__ void k_attn_node(const float* __restrict__ xl,
                            const float* __restrict__ att_src,
                            const float* __restrict__ att_dst,
                            float* __restrict__ a_s, float* __restrict__ a_d, int n) {
    const int lane = threadIdx.x & 31;
    const int row  = blockIdx.x * 8 + (threadIdx.x >> 5);
    if (row >= n) return;
    float x0 = xl[(long long)row * 64 + lane];
    float x1 = xl[(long long)row * 64 + 32 + lane];
    float ps = x0 * att_src[lane] + x1 * att_src[32 + lane];
    float pd = x0 * att_dst[lane] + x1 * att_dst[32 + lane];
    for (int off = 16; off > 0; off >>= 1) {
        ps += __shfl_down(ps, off);
        pd += __shfl_down(pd, off);
    }
    if (lane == 0) { a_s[row] = ps; a_d[row] = pd; }
}

// ---------- edge pass 1: leaky-relu logits + segment max (ordered-uint atomicMax) ----------
__global__ void k_edge1(const int* __restrict__ ei, const float* __restrict__ ea,
                        const float* __restrict__ a_s, const float* __restrict__ a_d,
                        const float* __restrict__ scal, int ce_idx,
                        float* __restrict__ ebuf, unsigned* __restrict__ mu,
                        int E, int N) {
    long long i = (long long)blockIdx.x * blockDim.x + threadIdx.x;
    if (i >= (long long)E + N) return;
    int s, d; float av;
    if (i < E) { s = ei[i]; d = ei[E + i]; av = ea[i]; }
    else       { s = d = (int)(i - E);     av = scal[0] / (float)E; }
    float e = a_s[s] + a_d[d] + av * scal[ce_idx];
    e = (e > 0.0f) ? e : NEG_SLOPE * e;
    ebuf[i] = e;
    atomicMax(&mu[d], fenc(e));
}

// ---------- edge pass 2: ex = exp(e - max), segment sum ----------
__global__ void k_edge2(const int* __restrict__ ei, float* __restrict__ ebuf,
                        const unsigned* __restrict__ mu, float* __restrict__ ssum,
                        int E, int N) {
    long long i = (long long)blockIdx.x * blockDim.x + threadIdx.x;
    if (i >= (long long)E + N) return;
    int d = (i < E) ? ei[E + i] : (int)(i - E);
    float ex = expf(ebuf[i] - fdec(mu[d]));
    ebuf[i] = ex;
    atomicAdd(&ssum[d], ex);
}

// ---------- edge pass 3: agg[d] += alpha * xl[s]  (64 lanes per edge) ----------
__global__ void k_edge3(const int* __restrict__ ei, const float* __restrict__ ebuf,
                        const float* __restrict__ ssum, const float* __restrict__ xl,
                        float* __restrict__ agg, int E, int N) {
    long long g = (long long)blockIdx.x * blockDim.x + threadIdx.x;
    long long i = g >> 6;
    int c = (int)(g & 63);
    if (i >= (long long)E + N) return;
    int s, d;
    if (i < E) { s = ei[i]; d = ei[E + i]; }
    else       { s = d = (int)(i - E); }
    float alpha = ebuf[i] / (ssum[d] + 1e-16f);
    atomicAdd(&agg[(long long)d * 64 + c], alpha * xl[(long long)s * 64 + c]);
}

// ---------- finish layer1: h = elu(agg + b1); reset m/s for layer 2 ----------
__global__ void k_finish1(const float* __restrict__ agg, const float* __restrict__ b1,
                          float* __restrict__ h, unsigned* __restrict__ mu,
                          float* __restrict__ ssum, long long n64, int n) {
    long long i = (long long)blockIdx.x * blockDim.x + threadIdx.x;
    if (i < n64) {
        float v = agg[i] + b1[i & 63];
        h[i] = (v > 0.0f) ? v : (expf(v) - 1.0f);
    }
    if (i < n) { mu[i] = 0u; ssum[i] = 0.0f; }
}

// ---------- final: out = agg2 + b2 + alpha_mix * x_res ----------
__global__ void k_final(float* __restrict__ out, const float* __restrict__ b2,
                        const float* __restrict__ amix, const float* __restrict__ xres,
                        long long n64) {
    long long i = (long long)blockIdx.x * blockDim.x + threadIdx.x;
    if (i >= n64) return;
    out[i] = out[i] + b2[i & 63] + amix[0] * xres[i];
}

extern "C" void kernel_launch(void* const* d_in, const int* in_sizes, int n_in,
                              void* d_out, int out_size, void* d_ws, size_t ws_size,
                              hipStream_t stream) {
    const float* x     = (const float*)d_in[0];
    const int*   ei    = (const int*)d_in[1];
    const float* ea    = (const float*)d_in[2];
    const float* W_res = (const float*)d_in[3];
    const float* b_res = (const float*)d_in[4];
    const float* amix  = (const float*)d_in[5];
    const float* W1    = (const float*)d_in[6];
    const float* as1   = (const float*)d_in[7];
    const float* ad1   = (const float*)d_in[8];
    const float* We1   = (const float*)d_in[9];
    const float* ae1   = (const float*)d_in[10];
    const float* b1    = (const float*)d_in[11];
    const float* W2    = (const float*)d_in[12];
    const float* as2   = (const float*)d_in[13];
    const float* ad2   = (const float*)d_in[14];
    const float* We2   = (const float*)d_in[15];
    const float* ae2   = (const float*)d_in[16];
    const float* b2    = (const float*)d_in[17];
    float* out = (float*)d_out;

    const int N = in_sizes[0] / 64;
    const int E = in_sizes[2];
    const long long total = (long long)E + N;
    const long long N64 = (long long)N * 64;

    float* p = (float*)d_ws;
    float* xres = p; p += N64;
    float* xl   = p; p += N64;   // xl1, then h after finish1
    float* agg  = p; p += N64;   // agg1, then xl2
    float* ebuf = p; p += total;
    float* a_s  = p; p += N;
    float* a_d  = p; p += N;
    float* ssum = p; p += N;
    unsigned* mu = (unsigned*)p; p += N;
    float* scal = p; p += 8;

    const int T = 256;
    const int gN64  = (int)((N64 + T - 1) / T);
    const int gEdge = (int)((total + T - 1) / T);
    const int gE3   = (int)((total * 64 + T - 1) / T);
    const int gAttn = (N + 7) / 8;
    const int rowTiles = (N + 15) / 16;
    const int gGemm = (rowTiles + 1) / 2;

    // init accumulators (d_out is layer-2 aggregation target)
    k_init<<<gN64, T, 0, stream>>>(out, agg, mu, ssum, scal, N64, N);
    // scalar reductions: mean(edge_attr) numerator, ce1, ce2
    k_reduce<<<512, T, 0, stream>>>(ea, E, We1, ae1, We2, ae2, scal);

    // residual: x_res = x @ W_res + b_res
    k_gemm64<true><<<gGemm, T, 0, stream>>>(x, W_res, b_res, xres, N);

    // ---- layer 1 ----
    k_gemm64<false><<<gGemm, T, 0, stream>>>(xres, W1, nullptr, xl, N);
    k_attn_node<<<gAttn, T, 0, stream>>>(xl, as1, ad1, a_s, a_d, N);
    k_edge1<<<gEdge, T, 0, stream>>>(ei, ea, a_s, a_d, scal, 1, ebuf, mu, E, N);
    k_edge2<<<gEdge, T, 0, stream>>>(ei, ebuf, mu, ssum, E, N);
    k_edge3<<<gE3, T, 0, stream>>>(ei, ebuf, ssum, xl, agg, E, N);
    // h = elu(agg1 + b1) -> xl buffer; reset segment state for layer 2
    k_finish1<<<gN64, T, 0, stream>>>(agg, b1, xl, mu, ssum, N64, N);

    // ---- layer 2 ----  (xl2 lives in agg buffer; aggregation goes into d_out)
    k_gemm64<false><<<gGemm, T, 0, stream>>>(xl, W2, nullptr, agg, N);
    k_attn_node<<<gAttn, T, 0, stream>>>(agg, as2, ad2, a_s, a_d, N);
    k_edge1<<<gEdge, T, 0, stream>>>(ei, ea, a_s, a_d, scal, 2, ebuf, mu, E, N);
    k_edge2<<<gEdge, T, 0, stream>>>(ei, ebuf, mu, ssum, E, N);
    k_edge3<<<gE3, T, 0, stream>>>(ei, ebuf, ssum, agg, out, E, N);

    // out = agg2 + b2 + alpha_mix * x_res
    k_final<<<gN64, T, 0, stream>>>(out, b2, amix, xres, N64);
}